// AssemblyArrayComponent_9019431322130
// MI455X (gfx1250) — compile-verified
//
#include <hip/hip_runtime.h>
#include <math.h>
#include <stdint.h>

typedef __attribute__((ext_vector_type(2))) float v2f;
typedef __attribute__((ext_vector_type(8))) float v8f;

// ---- problem constants ----
static constexpr int CB = 16;        // batch
static constexpr int CLC = 512;      // after conv2 (stride 4)
static constexpr int CL1 = 2048;     // after conv1 (stride 8)
static constexpr int ROWS1 = CB * CL1;   // 32768
static constexpr int ROWS  = CB * CLC;   // 8192
static constexpr int CFF = 256;
static constexpr int D_STATE = 32, D_INNER = 256, NHEADS = 8;
static constexpr int CONV_DIM = 320;     // D_INNER + 2*D_STATE
static constexpr int NPROJ = 584;        // 2*D_INNER + 2*D_STATE + NHEADS

__device__ __forceinline__ float gelu_t(float x) {
  float x3 = x * x * x;
  return 0.5f * x * (1.0f + tanhf(0.7978845608028654f * (x + 0.044715f * x3)));
}
__device__ __forceinline__ float silu_f(float x) { return x / (1.0f + expf(-x)); }

// async 16B global->LDS copy (gfx1250 GLOBAL_LOAD_ASYNC_TO_LDS_B128, ASYNCcnt)
__device__ __forceinline__ void async_b128(uint32_t lds_off, const float* g) {
  uint64_t ga = (uint64_t)g;
  asm volatile("global_load_async_to_lds_b128 %0, %1, off"
               :: "v"(lds_off), "v"(ga) : "memory");
}
#define WAIT_ASYNC(n) asm volatile("s_wait_asynccnt %0" :: "i"(n) : "memory")

// ======================= WMMA fp32 GEMM with async-LDS staging ==========
// C[M,N] = A[M,K] @ B[K,N] (+epilogue).  Block = 128 threads (4 waves).
// Tile: 64(M) x 16(N); K consumed in 32-wide chunks, double-buffered LDS.
// Requires: M % 64 == 0, K % 32 == 0, N % 4 == 0 (true for all calls here).
// MODE 0: none
// MODE 1: gelu(acc*p0[col] + p1[col])
// MODE 2: elu(acc)+1
// MODE 3: acc + p0[col] + res[row,col]
// MODE 4: gelu(acc + p0[col])
static constexpr int TK = 32;
static constexpr int APAD = 33;   // A-tile row stride in dwords (bank-conflict pad)

template <int MODE>
__global__ __launch_bounds__(128) void gemm_wmma(
    const float* __restrict__ A, const float* __restrict__ Bm,
    float* __restrict__ C, const float* __restrict__ p0,
    const float* __restrict__ p1, const float* __restrict__ res,
    int M, int N, int K) {
  __shared__ float As[2][64 * APAD];
  __shared__ float Bs[2][TK * 16];
  const int tid = threadIdx.y * 32 + threadIdx.x;
  const int lane = threadIdx.x;
  const int half = lane >> 4;
  const int m = lane & 15;
  const int wv = threadIdx.y;
  const int tm0 = blockIdx.y * 64;
  const int tn = blockIdx.x * 16;
  const int bcol = tn + m;
  const float bmask = (bcol < N) ? 1.0f : 0.0f;
  const int lrow = wv * 16 + m;

  const uint32_t asbase = (uint32_t)(uintptr_t)(&As[0][0]);
  const uint32_t bsbase = (uint32_t)(uintptr_t)(&Bs[0][0]);

  // issue one K-chunk (5 async b128 per thread: 4 for A, 1 for B)
  auto issue_chunk = [&](int k0, int buf) {
#pragma unroll
    for (int i = 0; i < 4; ++i) {
      int id = tid + i * 128;            // 512 lane-copies: 64 rows x 8 segs
      int row = id >> 3, seg = id & 7;
      const float* g = A + (size_t)(tm0 + row) * K + k0 + seg * 4;
      uint32_t l = asbase + (uint32_t)(buf * (64 * APAD) + row * APAD + seg * 4) * 4u;
      async_b128(l, g);
    }
    {
      int kr = tid >> 2, seg = tid & 3;  // 128 lane-copies: 32 rows x 4 segs
      int gc = tn + seg * 4;
      if (gc + 4 > N) gc = N - 4;        // N%4==0 -> clamped segs fully masked at use
      const float* g = Bm + (size_t)(k0 + kr) * N + gc;
      uint32_t l = bsbase + (uint32_t)(buf * (TK * 16) + kr * 16 + seg * 4) * 4u;
      async_b128(l, g);
    }
  };

  v8f acc = {};
  const int nc = K / TK;
  issue_chunk(0, 0);
  for (int c = 0; c < nc; ++c) {
    if (c + 1 < nc) {
      issue_chunk((c + 1) * TK, (c + 1) & 1);
      WAIT_ASYNC(5);   // in-order completion: <=5 outstanding => chunk c landed
    } else {
      WAIT_ASYNC(0);
    }
    __syncthreads();
    const int buf = c & 1;
#pragma unroll
    for (int kkl = 0; kkl < TK; kkl += 4) {
      const int ka = kkl + 2 * half;
      v2f av, bv;
      av.x = As[buf][lrow * APAD + ka];
      av.y = As[buf][lrow * APAD + ka + 1];
      bv.x = Bs[buf][ka * 16 + m] * bmask;
      bv.y = Bs[buf][(ka + 1) * 16 + m] * bmask;
      acc = __builtin_amdgcn_wmma_f32_16x16x4_f32(false, av, false, bv,
                                                  (short)0, acc, false, false);
    }
    __syncthreads();   // protect buf from being overwritten by chunk c+2
  }

  if (bcol < N) {
    const int tm = tm0 + wv * 16;
#pragma unroll
    for (int v = 0; v < 8; ++v) {
      const int row = tm + v + 8 * half;
      float x = acc[v];
      if constexpr (MODE == 1) x = gelu_t(x * p0[bcol] + p1[bcol]);
      else if constexpr (MODE == 2) x = (x > 0.0f) ? x + 1.0f : expf(x);
      else if constexpr (MODE == 3) x = x + p0[bcol] + res[(size_t)row * N + bcol];
      else if constexpr (MODE == 4) x = gelu_t(x + p0[bcol]);
      C[(size_t)row * N + bcol] = x;
    }
  }
}

// ======================= prep kernels =======================
__global__ void prep_weff(const float* __restrict__ w_in,
                          const float* __restrict__ conv1_w,
                          float* __restrict__ Weff) {
  int idx = blockIdx.x * blockDim.x + threadIdx.x;  // 16384
  int j = idx >> 7, co = idx & 127;
  int k = j >> 4, e = j & 15;
  float s = 0.0f;
  for (int ci = 0; ci < 128; ++ci)
    s += w_in[e * 128 + ci] * conv1_w[co * 1024 + ci * 8 + k];
  Weff[j * 128 + co] = s;
}

__global__ void prep_scales(const float* __restrict__ b_in,
                            const float* __restrict__ conv1_w,
                            const float* __restrict__ bn1_g, const float* __restrict__ bn1_b,
                            const float* __restrict__ bn2_g, const float* __restrict__ bn2_b,
                            float* __restrict__ s1, float* __restrict__ t1,
                            float* __restrict__ s2, float* __restrict__ t2) {
  int co = threadIdx.x;
  float inv = rsqrtf(1.0f + 1e-5f);
  float be = 0.0f;
  for (int k = 0; k < 8; ++k)
    for (int ci = 0; ci < 128; ++ci)
      be += b_in[ci] * conv1_w[co * 1024 + ci * 8 + k];
  float sc = bn1_g[co] * inv;
  s1[co] = sc;
  t1[co] = be * sc + bn1_b[co];
  s2[co] = bn2_g[co] * inv;
  t2[co] = bn2_b[co];
}

__global__ void prep_w2(const float* __restrict__ conv2_w, float* __restrict__ W2p) {
  int idx = blockIdx.x * blockDim.x + threadIdx.x;  // 65536
  int j = idx >> 7, co = idx & 127;
  int k = j >> 7, ci = j & 127;
  W2p[j * 128 + co] = conv2_w[co * 512 + ci * 4 + k];
}

// ======================= row norms =======================
__device__ __forceinline__ float block_reduce_sum(float val, float* sm) {
  int tid = threadIdx.x;
  sm[tid] = val;
  __syncthreads();
  for (int s = blockDim.x >> 1; s > 0; s >>= 1) {
    if (tid < s) sm[tid] += sm[tid + s];
    __syncthreads();
  }
  float r = sm[0];
  __syncthreads();
  return r;
}

__global__ __launch_bounds__(128) void ln_kernel(const float* __restrict__ in,
                                                 const float* __restrict__ g,
                                                 const float* __restrict__ b,
                                                 float* __restrict__ out) {
  __shared__ float sm[128];
  int row = blockIdx.x, c = threadIdx.x;
  float x = in[(size_t)row * 128 + c];
  float mean = block_reduce_sum(x, sm) * (1.0f / 128.0f);
  float d = x - mean;
  float var = block_reduce_sum(d * d, sm) * (1.0f / 128.0f);
  out[(size_t)row * 128 + c] = d * rsqrtf(var + 1e-5f) * g[c] + b[c];
}

// ======================= linear attention =======================
__global__ void ksum_kernel(const float* __restrict__ kb, float* __restrict__ ksum) {
  int idx = blockIdx.x * blockDim.x + threadIdx.x;  // 16*128
  if (idx >= CB * 128) return;
  int b = idx >> 7, hd = idx & 127;
  const float* p = kb + (size_t)(b * CLC) * 128 + hd;
  float s = 0.0f;
  for (int l = 0; l < CLC; ++l) s += p[(size_t)l * 128];
  ksum[idx] = s;
}

// kv[b,h][d][e] = sum_l k[b,l,h,d]*v[b,l,h,e]; 4 waves = four 16x16 tiles
__global__ __launch_bounds__(128) void kv_wmma(const float* __restrict__ kq,
                                               const float* __restrict__ vq,
                                               float* __restrict__ kvb) {
  const int b = blockIdx.x >> 2, h = blockIdx.x & 3;
  const int wv = threadIdx.y;
  const int tmd = (wv >> 1) * 16, tne = (wv & 1) * 16;
  const int lane = threadIdx.x, half = lane >> 4, m = lane & 15;
  const size_t base = (size_t)(b * CLC) * 128 + h * 32;
  v8f acc = {};
#pragma unroll 4
  for (int kk = 0; kk < CLC; kk += 4) {
    const int ka = kk + 2 * half;
    v2f av, bv;
    av.x = kq[base + (size_t)ka * 128 + tmd + m];
    av.y = kq[base + (size_t)(ka + 1) * 128 + tmd + m];
    bv.x = vq[base + (size_t)ka * 128 + tne + m];
    bv.y = vq[base + (size_t)(ka + 1) * 128 + tne + m];
    acc = __builtin_amdgcn_wmma_f32_16x16x4_f32(false, av, false, bv,
                                                (short)0, acc, false, false);
  }
  float* out = kvb + (size_t)(b * 4 + h) * 1024;
#pragma unroll
  for (int v = 0; v < 8; ++v)
    out[(tmd + v + 8 * half) * 32 + tne + m] = acc[v];
}

__global__ __launch_bounds__(128) void attn_kernel(const float* __restrict__ qb,
                                                   const float* __restrict__ kvb,
                                                   const float* __restrict__ ksum,
                                                   float* __restrict__ attn) {
  __shared__ float qs[128];
  int row = blockIdx.x, col = threadIdx.x;
  int b = row >> 9;
  int hh = col >> 5, e = col & 31;
  qs[col] = qb[(size_t)row * 128 + col];
  __syncthreads();
  const float* kvp = kvb + (size_t)(b * 4 + hh) * 1024 + e;
  const float* ksp = ksum + b * 128 + hh * 32;
  float num = 0.0f, den = 0.0f;
#pragma unroll
  for (int d = 0; d < 32; ++d) {
    float qd = qs[hh * 32 + d];
    num += qd * kvp[d * 32];
    den += qd * ksp[d];
  }
  attn[(size_t)row * 128 + col] = num / (den + 1e-6f);
}

// ======================= mamba block =======================
__global__ void dtda_kernel(const float* __restrict__ zx,
                            const float* __restrict__ dt_bias,
                            const float* __restrict__ A_log,
                            float* __restrict__ dtb, float* __restrict__ dAb) {
  int idx = blockIdx.x * blockDim.x + threadIdx.x;  // 8192*8
  if (idx >= ROWS * NHEADS) return;
  int row = idx >> 3, i = idx & 7;
  float v = zx[(size_t)row * NPROJ + 2 * D_INNER + 2 * D_STATE + i] + dt_bias[i];
  float dt = (v > 20.0f) ? v : log1pf(expf(v));
  dtb[idx] = dt;
  dAb[idx] = expf(dt * (-expf(A_log[i])));
}

__global__ void dwconv_kernel(const float* __restrict__ zx,
                              const float* __restrict__ conv_w,
                              const float* __restrict__ conv_b,
                              float* __restrict__ xc) {
  int idx = blockIdx.x * blockDim.x + threadIdx.x;  // 8192*320
  if (idx >= ROWS * CONV_DIM) return;
  int row = idx / CONV_DIM, c = idx % CONV_DIM;
  int b = row >> 9, t = row & 511;
  float acc = conv_b[c];
#pragma unroll
  for (int i = 0; i < 4; ++i) {
    int tt = t - 3 + i;
    if (tt >= 0)
      acc += zx[(size_t)((b << 9) + tt) * NPROJ + D_INNER + c] * conv_w[c * 4 + i];
  }
  xc[idx] = silu_f(acc);
}

// selective scan: one wave per (b, head); lane owns state row p (32 regs)
__global__ __launch_bounds__(32) void scan_kernel(const float* __restrict__ xc,
                                                  const float* __restrict__ dtb,
                                                  const float* __restrict__ dAb,
                                                  float* __restrict__ ys) {
  int b = blockIdx.x >> 3, h = blockIdx.x & 7;
  int p = threadIdx.x;
  float state[32];
#pragma unroll
  for (int n = 0; n < 32; ++n) state[n] = 0.0f;
  for (int t = 0; t < CLC; ++t) {
    int row = (b << 9) + t;
    float dAv = dAb[row * 8 + h];
    float dtv = dtb[row * 8 + h];
    const float* xr = xc + (size_t)row * CONV_DIM;
    float xsp = xr[h * 32 + p];
    float Bn = xr[D_INNER + p];
    float Cn = xr[D_INNER + D_STATE + p];
    float coeff = dtv * xsp;
    float acc = 0.0f;
#pragma unroll
    for (int n = 0; n < 32; ++n) {
      float bb = __shfl(Bn, n, 32);
      float cc = __shfl(Cn, n, 32);
      state[n] = state[n] * dAv + coeff * bb;
      acc += state[n] * cc;
    }
    ys[(size_t)row * D_INNER + h * 32 + p] = acc;
  }
}

__global__ __launch_bounds__(256) void gate_rms_kernel(const float* __restrict__ ys,
                                                       const float* __restrict__ xc,
                                                       const float* __restrict__ zx,
                                                       const float* __restrict__ D_skip,
                                                       const float* __restrict__ mnorm_w,
                                                       float* __restrict__ yn) {
  __shared__ float sm[256];
  int row = blockIdx.x, c = threadIdx.x;
  float yv = ys[(size_t)row * D_INNER + c];
  float xsv = xc[(size_t)row * CONV_DIM + c];
  float zgv = zx[(size_t)row * NPROJ + c];
  float y2 = (yv + D_skip[c >> 5] * xsv) * silu_f(zgv);
  float ss = block_reduce_sum(y2 * y2, sm);
  float scale = rsqrtf(ss * (1.0f / 256.0f) + 1e-6f);
  yn[(size_t)row * D_INNER + c] = y2 * scale * mnorm_w[c];
}

__global__ __launch_bounds__(128) void final_kernel(const float* __restrict__ ot,
                                                    const float* __restrict__ h2,
                                                    const float* __restrict__ rms_w,
                                                    const float* __restrict__ oln_g,
                                                    const float* __restrict__ oln_b,
                                                    float* __restrict__ out) {
  __shared__ float sm[128];
  int row = blockIdx.x, c = threadIdx.x;
  float u = ot[(size_t)row * 128 + c] + h2[(size_t)row * 128 + c];
  float ss = block_reduce_sum(u * u, sm);
  float r = u * rsqrtf(ss * (1.0f / 128.0f) + 1e-6f) * rms_w[c];
  float mean = block_reduce_sum(r, sm) * (1.0f / 128.0f);
  float d = r - mean;
  float var = block_reduce_sum(d * d, sm) * (1.0f / 128.0f);
  out[(size_t)row * 128 + c] = d * rsqrtf(var + 1e-5f) * oln_g[c] + oln_b[c];
}

// ======================= launcher =======================
static void launch_gemm(int mode, const float* A, const float* Bm, float* C,
                        const float* p0, const float* p1, const float* res,
                        int M, int N, int K, hipStream_t s) {
  dim3 blk(32, 4);
  dim3 grd((N + 15) / 16, M / 64);
  switch (mode) {
    case 0: gemm_wmma<0><<<grd, blk, 0, s>>>(A, Bm, C, p0, p1, res, M, N, K); break;
    case 1: gemm_wmma<1><<<grd, blk, 0, s>>>(A, Bm, C, p0, p1, res, M, N, K); break;
    case 2: gemm_wmma<2><<<grd, blk, 0, s>>>(A, Bm, C, p0, p1, res, M, N, K); break;
    case 3: gemm_wmma<3><<<grd, blk, 0, s>>>(A, Bm, C, p0, p1, res, M, N, K); break;
    case 4: gemm_wmma<4><<<grd, blk, 0, s>>>(A, Bm, C, p0, p1, res, M, N, K); break;
  }
}

extern "C" void kernel_launch(void* const* d_in, const int* in_sizes, int n_in,
                              void* d_out, int out_size, void* d_ws, size_t ws_size,
                              hipStream_t stream) {
  const float* x        = (const float*)d_in[0];
  const float* w_in     = (const float*)d_in[1];
  const float* b_in     = (const float*)d_in[2];
  const float* conv1_w  = (const float*)d_in[3];
  const float* bn1_g    = (const float*)d_in[4];
  const float* bn1_b    = (const float*)d_in[5];
  const float* conv2_w  = (const float*)d_in[6];
  const float* bn2_g    = (const float*)d_in[7];
  const float* bn2_b    = (const float*)d_in[8];
  const float* ln1_g    = (const float*)d_in[9];
  const float* ln1_b    = (const float*)d_in[10];
  const float* wq       = (const float*)d_in[11];
  const float* wk       = (const float*)d_in[12];
  const float* wv       = (const float*)d_in[13];
  const float* wo       = (const float*)d_in[14];
  const float* bo       = (const float*)d_in[15];
  const float* ln2_g    = (const float*)d_in[16];
  const float* ln2_b    = (const float*)d_in[17];
  const float* ff1_w    = (const float*)d_in[18];
  const float* ff1_b    = (const float*)d_in[19];
  const float* ff2_w    = (const float*)d_in[20];
  const float* ff2_b    = (const float*)d_in[21];
  const float* in_proj_w= (const float*)d_in[22];
  const float* conv_w   = (const float*)d_in[23];
  const float* conv_b   = (const float*)d_in[24];
  const float* dt_bias  = (const float*)d_in[25];
  const float* A_log    = (const float*)d_in[26];
  const float* D_skip   = (const float*)d_in[27];
  const float* mnorm_w  = (const float*)d_in[28];
  const float* out_w    = (const float*)d_in[29];
  const float* rms_w    = (const float*)d_in[30];
  const float* oln_g    = (const float*)d_in[31];
  const float* oln_b    = (const float*)d_in[32];
  float* out = (float*)d_out;

  float* ws = (float*)d_ws;
  size_t off = 0;
  auto alloc = [&](size_t n) { float* p = ws + off; off += n; return p; };
  float* Weff  = alloc(128 * 128);
  float* W2p   = alloc(512 * 128);
  float* s1    = alloc(128);
  float* t1    = alloc(128);
  float* s2    = alloc(128);
  float* t2    = alloc(128);
  float* g1    = alloc((size_t)ROWS1 * 128);
  float* h0    = alloc((size_t)ROWS * 128);
  float* abuf  = alloc((size_t)ROWS * 128);
  float* qb    = alloc((size_t)ROWS * 128);
  float* kb    = alloc((size_t)ROWS * 128);
  float* vb    = alloc((size_t)ROWS * 128);
  float* ksumb = alloc(CB * 128);
  float* kvb   = alloc(CB * 4 * 1024);
  float* attn  = alloc((size_t)ROWS * 128);
  float* h1    = alloc((size_t)ROWS * 128);
  float* fb    = alloc((size_t)ROWS * 128);
  float* ffh   = alloc((size_t)ROWS * CFF);
  float* h2    = alloc((size_t)ROWS * 128);
  float* zx    = alloc((size_t)ROWS * NPROJ);
  float* dtb   = alloc((size_t)ROWS * NHEADS);
  float* dAb   = alloc((size_t)ROWS * NHEADS);
  float* xcb   = alloc((size_t)ROWS * CONV_DIM);
  float* ysb   = alloc((size_t)ROWS * D_INNER);
  float* ynb   = alloc((size_t)ROWS * D_INNER);
  float* otb   = alloc((size_t)ROWS * 128);
  (void)ws_size; (void)in_sizes; (void)n_in; (void)out_size;

  // prep: fold w_in into conv1, pack conv2 as GEMM-B, bn scale/shift
  prep_weff<<<64, 256, 0, stream>>>(w_in, conv1_w, Weff);
  prep_scales<<<1, 128, 0, stream>>>(b_in, conv1_w, bn1_g, bn1_b, bn2_g, bn2_b, s1, t1, s2, t2);
  prep_w2<<<256, 256, 0, stream>>>(conv2_w, W2p);

  // stem: x(as 32768x128 patches) @ Weff -> gelu(bn1); patches @ W2p -> gelu(bn2)
  launch_gemm(1, x, Weff, g1, s1, t1, nullptr, ROWS1, 128, 128, stream);
  launch_gemm(1, g1, W2p, h0, s2, t2, nullptr, ROWS, 128, 512, stream);

  // linear attention
  ln_kernel<<<ROWS, 128, 0, stream>>>(h0, ln1_g, ln1_b, abuf);
  launch_gemm(2, abuf, wq, qb, nullptr, nullptr, nullptr, ROWS, 128, 128, stream);
  launch_gemm(2, abuf, wk, kb, nullptr, nullptr, nullptr, ROWS, 128, 128, stream);
  launch_gemm(0, abuf, wv, vb, nullptr, nullptr, nullptr, ROWS, 128, 128, stream);
  ksum_kernel<<<(CB * 128 + 255) / 256, 256, 0, stream>>>(kb, ksumb);
  kv_wmma<<<CB * 4, dim3(32, 4), 0, stream>>>(kb, vb, kvb);
  attn_kernel<<<ROWS, 128, 0, stream>>>(qb, kvb, ksumb, attn);
  launch_gemm(3, attn, wo, h1, bo, nullptr, h0, ROWS, 128, 128, stream);

  // FFN
  ln_kernel<<<ROWS, 128, 0, stream>>>(h1, ln2_g, ln2_b, fb);
  launch_gemm(4, fb, ff1_w, ffh, ff1_b, nullptr, nullptr, ROWS, CFF, 128, stream);
  launch_gemm(3, ffh, ff2_w, h2, ff2_b, nullptr, h1, ROWS, 128, CFF, stream);

  // mamba block
  launch_gemm(0, h2, in_proj_w, zx, nullptr, nullptr, nullptr, ROWS, NPROJ, 128, stream);
  dtda_kernel<<<(ROWS * NHEADS + 255) / 256, 256, 0, stream>>>(zx, dt_bias, A_log, dtb, dAb);
  dwconv_kernel<<<(ROWS * CONV_DIM + 255) / 256, 256, 0, stream>>>(zx, conv_w, conv_b, xcb);
  scan_kernel<<<CB * NHEADS, 32, 0, stream>>>(xcb, dtb, dAb, ysb);
  gate_rms_kernel<<<ROWS, 256, 0, stream>>>(ysb, xcb, zx, D_skip, mnorm_w, ynb);

  // output projection + final norms
  launch_gemm(0, ynb, out_w, otb, nullptr, nullptr, nullptr, ROWS, 128, D_INNER, stream);
  final_kernel<<<ROWS, 128, 0, stream>>>(otb, h2, rms_w, oln_g, oln_b, out);
}